// GATModel_12695923327312
// MI455X (gfx1250) — compile-verified
//
#include <hip/hip_runtime.h>

typedef __attribute__((ext_vector_type(2))) float v2f;
typedef __attribute__((ext_vector_type(8))) float v8f;

#define NEG_SLOPE 0.2f

// ---- order-preserving float -> uint key for atomicMax on floats ----
__device__ __forceinline__ unsigned fmax_key(float f) {
  unsigned u = __float_as_uint(f);
  return (u & 0x80000000u) ? ~u : (u | 0x80000000u);
}
__device__ __forceinline__ float fmax_unkey(unsigned k) {
  return (k & 0x80000000u) ? __uint_as_float(k & 0x7fffffffu) : __uint_as_float(~k);
}

__global__ void k_zero(float* __restrict__ p, long n) {
  long i = (long)blockIdx.x * blockDim.x + threadIdx.x;
  if (i < n) p[i] = 0.f;
}

// xw = x[N,4] @ W[4,64] via V_WMMA_F32_16X16X4_F32 (K=4 exact, 4 N-tiles of 16).
// Tail-safe without branches: A-rows >= n are clamp-loaded from row n-1, so their
// outputs equal row n-1's; storing them to min(m, n-1) is a benign same-value race.
__global__ void k_lin1(const float* __restrict__ x, const float* __restrict__ W,
                       float* __restrict__ xw, int n) {
  int wave = (int)((blockIdx.x * blockDim.x + threadIdx.x) >> 5);
  int lane = threadIdx.x & 31;
  int half = lane >> 4;        // K split: lanes 0-15 -> K=0,1 ; lanes 16-31 -> K=2,3
  int lr   = lane & 15;
  int base = wave * 16;
  if (base >= n) return;       // wave-uniform exit
  int row = min(base + lr, n - 1);
  int m0  = base + half * 8;   // C layout: VGPR r -> row M=r (lanes 0-15) / M=8+r (lanes 16-31)
  v2f a;
  a.x = x[row * 4 + 2 * half + 0];
  a.y = x[row * 4 + 2 * half + 1];
  #pragma unroll
  for (int t = 0; t < 4; ++t) {
    int col = t * 16 + lr;
    v2f b;
    b.x = W[(2 * half + 0) * 64 + col];
    b.y = W[(2 * half + 1) * 64 + col];
    v8f c = {};
    c = __builtin_amdgcn_wmma_f32_16x16x4_f32(false, a, false, b, (short)0, c, false, false);
    #pragma unroll
    for (int r = 0; r < 8; ++r) {
      int m = min(m0 + r, n - 1);
      xw[(size_t)m * 64 + t * 16 + lr] = c[r];
    }
  }
}

// xw = h[N,64] @ W[64,8] via 16 chained V_WMMA_F32_16X16X4_F32 (cols 8..15 zero-padded)
__global__ void k_lin2(const float* __restrict__ h, const float* __restrict__ W,
                       float* __restrict__ xw, int n) {
  int wave = (int)((blockIdx.x * blockDim.x + threadIdx.x) >> 5);
  int lane = threadIdx.x & 31;
  int half = lane >> 4;
  int lr   = lane & 15;
  int base = wave * 16;
  if (base >= n) return;       // wave-uniform exit
  int row = min(base + lr, n - 1);
  int m0  = base + half * 8;
  v8f c = {};
  #pragma unroll
  for (int k = 0; k < 16; ++k) {
    int kk = 4 * k + 2 * half;
    v2f a;
    a.x = h[(size_t)row * 64 + kk + 0];
    a.y = h[(size_t)row * 64 + kk + 1];
    v2f b;
    b.x = (lr < 8) ? W[(kk + 0) * 8 + lr] : 0.f;
    b.y = (lr < 8) ? W[(kk + 1) * 8 + lr] : 0.f;
    c = __builtin_amdgcn_wmma_f32_16x16x4_f32(false, a, false, b, (short)0, c, false, false);
  }
  if (lr < 8) {                // single divergent guard: padding columns must not store
    #pragma unroll
    for (int r = 0; r < 8; ++r) {
      int m = min(m0 + r, n - 1);
      xw[(size_t)m * 8 + lr] = c[r];
    }
  }
}

// per-(node,head) attention logit halves: as = <xw[n,h,:], att_src[h,:]>, same for ad
template <int H, int C>
__global__ void k_alpha(const float* __restrict__ xw, const float* __restrict__ att_s,
                        const float* __restrict__ att_d, float* __restrict__ as_,
                        float* __restrict__ ad_, int n) {
  int i = blockIdx.x * blockDim.x + threadIdx.x;
  if (i >= n * H) return;
  int node = i / H, h = i % H;
  const float* row = xw + (size_t)node * H * C + (size_t)h * C;
  float ss = 0.f, sd = 0.f;
  #pragma unroll
  for (int c = 0; c < C; ++c) {
    ss += row[c] * att_s[h * C + c];
    sd += row[c] * att_d[h * C + c];
  }
  as_[i] = ss; ad_[i] = sd;
}

// pass 1: segment max over incoming edges (incl. self-loops appended logically)
template <int H>
__global__ void k_edge_max(const int* __restrict__ ei, int E, int n,
                           const float* __restrict__ as_, const float* __restrict__ ad_,
                           unsigned* __restrict__ m_) {
  long e = (long)blockIdx.x * blockDim.x + threadIdx.x;
  long EA = (long)E + n;
  if (e >= EA) return;
  int src, dst;
  if (e < E) {
    __builtin_prefetch(&ei[e + 8192], 0, 0);               // stream ahead (edge list is HBM-streaming)
    __builtin_prefetch(&ei[(long)E + e + 8192], 0, 0);
    src = ei[e]; dst = ei[(long)E + e];
  } else {
    src = dst = (int)(e - E);
  }
  #pragma unroll
  for (int h = 0; h < H; ++h) {
    float s  = as_[src * H + h] + ad_[dst * H + h];
    float lv = s > 0.f ? s : NEG_SLOPE * s;
    atomicMax(&m_[dst * H + h], fmax_key(lv));
  }
}

// pass 2: w = exp(e - max); accumulate denom and unnormalized messages
template <int H, int C>
__global__ void k_edge_acc(const int* __restrict__ ei, int E, int n,
                           const float* __restrict__ xw,
                           const float* __restrict__ as_, const float* __restrict__ ad_,
                           const unsigned* __restrict__ m_, float* __restrict__ dn_,
                           float* __restrict__ acc) {
  long e = (long)blockIdx.x * blockDim.x + threadIdx.x;
  long EA = (long)E + n;
  if (e >= EA) return;
  int src, dst;
  if (e < E) {
    __builtin_prefetch(&ei[e + 8192], 0, 0);
    __builtin_prefetch(&ei[(long)E + e + 8192], 0, 0);
    src = ei[e]; dst = ei[(long)E + e];
  } else {
    src = dst = (int)(e - E);
  }
  #pragma unroll
  for (int h = 0; h < H; ++h) {
    float s  = as_[src * H + h] + ad_[dst * H + h];
    float lv = s > 0.f ? s : NEG_SLOPE * s;
    float mm = fmax_unkey(m_[dst * H + h]);
    float w  = __expf(lv - mm);
    atomicAdd(&dn_[dst * H + h], w);
    const float* srow = xw  + (size_t)src * H * C + (size_t)h * C;
    float*       drow = acc + (size_t)dst * H * C + (size_t)h * C;
    if constexpr ((C % 4) == 0) {
      // vectorized gather: b128 loads from the L2-resident feature table
      const float4* s4 = reinterpret_cast<const float4*>(srow);
      #pragma unroll
      for (int q = 0; q < C / 4; ++q) {
        float4 v = s4[q];
        atomicAdd(&drow[q * 4 + 0], w * v.x);
        atomicAdd(&drow[q * 4 + 1], w * v.y);
        atomicAdd(&drow[q * 4 + 2], w * v.z);
        atomicAdd(&drow[q * 4 + 3], w * v.w);
      }
    } else {
      #pragma unroll
      for (int c = 0; c < C; ++c) atomicAdd(&drow[c], w * srow[c]);
    }
  }
}

// normalize by denom, add bias, ELU (in place)
template <int H, int C>
__global__ void k_norm(float* __restrict__ acc, const float* __restrict__ dn_,
                       const float* __restrict__ bias, int n) {
  constexpr int HC = H * C;
  int i = blockIdx.x * blockDim.x + threadIdx.x;
  if (i >= n * HC) return;
  int node = i / HC;
  int h    = (i / C) % H;
  float v = acc[i] / (dn_[node * H + h] + 1e-16f) + bias[i % HC];
  acc[i] = v > 0.f ? v : expm1f(v);
}

// global mean pool into 8 accumulators
__global__ void k_pool(const float* __restrict__ h2, float* __restrict__ pool, int n) {
  int i = blockIdx.x * blockDim.x + threadIdx.x;
  if (i >= n) return;
  float inv = 1.0f / (float)n;
  const float4* r4 = reinterpret_cast<const float4*>(h2 + (size_t)i * 8);
  float4 a = r4[0], b = r4[1];
  atomicAdd(&pool[0], a.x * inv); atomicAdd(&pool[1], a.y * inv);
  atomicAdd(&pool[2], a.z * inv); atomicAdd(&pool[3], a.w * inv);
  atomicAdd(&pool[4], b.x * inv); atomicAdd(&pool[5], b.y * inv);
  atomicAdd(&pool[6], b.z * inv); atomicAdd(&pool[7], b.w * inv);
}

__global__ void k_final(const float* __restrict__ pool, const float* __restrict__ Wout,
                        const float* __restrict__ bout, float* __restrict__ out) {
  if (blockIdx.x == 0 && threadIdx.x == 0) {
    float z = bout[0];
    #pragma unroll
    for (int c = 0; c < 8; ++c) z += pool[c] * Wout[c];
    out[0] = 1.f / (1.f + __expf(-z));
  }
}

extern "C" void kernel_launch(void* const* d_in, const int* in_sizes, int n_in,
                              void* d_out, int out_size, void* d_ws, size_t ws_size,
                              hipStream_t stream) {
  const float* x    = (const float*)d_in[0];
  const int*   ei   = (const int*)d_in[1];   // JAX default disables x64 -> int32
  const float* W1   = (const float*)d_in[2];
  const float* as1  = (const float*)d_in[3];
  const float* ad1  = (const float*)d_in[4];
  const float* b1   = (const float*)d_in[5];
  const float* W2   = (const float*)d_in[6];
  const float* as2  = (const float*)d_in[7];
  const float* ad2  = (const float*)d_in[8];
  const float* b2   = (const float*)d_in[9];
  const float* Wout = (const float*)d_in[10];
  const float* bout = (const float*)d_in[11];
  float* out = (float*)d_out;

  int N = in_sizes[0] / 4;
  int E = in_sizes[1] / 2;

  // workspace carve-up (floats); region A reused across layers
  float*    regA = (float*)d_ws;                     // layer1: xw1 [N*64] ; layer2: xw2 [N*8] + h2 [N*8]
  float*    h1   = regA + (size_t)N * 64;            // [N*64] accumulator -> layer1 output
  float*    as_  = h1 + (size_t)N * 64;              // [N*4]
  float*    ad_  = as_ + (size_t)N * 4;              // [N*4]
  unsigned* m_   = (unsigned*)(ad_ + (size_t)N * 4); // [N*4] max keys
  float*    dn_  = (float*)(m_ + (size_t)N * 4);     // [N*4] denominators
  float*    pool = dn_ + (size_t)N * 4;              // [8]
  float*    xw2  = regA;
  float*    h2   = regA + (size_t)N * 8;

  long EA = (long)E + N;
  const int TB = 256;
  dim3 blk(TB);
  auto g = [&](long cnt) { return dim3((unsigned)((cnt + TB - 1) / TB)); };
  int waveThreads = ((N + 15) / 16) * 32;  // one wave per 16-node tile

  // ---------------- layer 1 ----------------
  k_zero<<<g((long)N * 64), blk, 0, stream>>>(h1, (long)N * 64);
  k_zero<<<g((long)N * 4),  blk, 0, stream>>>((float*)m_, (long)N * 4);
  k_zero<<<g((long)N * 4),  blk, 0, stream>>>(dn_, (long)N * 4);
  k_lin1<<<g(waveThreads), blk, 0, stream>>>(x, W1, regA, N);
  k_alpha<4, 16><<<g((long)N * 4), blk, 0, stream>>>(regA, as1, ad1, as_, ad_, N);
  k_edge_max<4><<<g(EA), blk, 0, stream>>>(ei, E, N, as_, ad_, m_);
  k_edge_acc<4, 16><<<g(EA), blk, 0, stream>>>(ei, E, N, regA, as_, ad_, m_, dn_, h1);
  k_norm<4, 16><<<g((long)N * 64), blk, 0, stream>>>(h1, dn_, b1, N);

  // ---------------- layer 2 ----------------
  k_zero<<<g((long)N * 4), blk, 0, stream>>>((float*)m_, (long)N * 4);
  k_zero<<<g((long)N * 4), blk, 0, stream>>>(dn_, (long)N * 4);
  k_zero<<<g((long)N * 8), blk, 0, stream>>>(h2, (long)N * 8);
  k_zero<<<dim3(1), blk, 0, stream>>>(pool, 8);
  k_lin2<<<g(waveThreads), blk, 0, stream>>>(h1, W2, xw2, N);
  k_alpha<4, 2><<<g((long)N * 4), blk, 0, stream>>>(xw2, as2, ad2, as_, ad_, N);
  k_edge_max<4><<<g(EA), blk, 0, stream>>>(ei, E, N, as_, ad_, m_);
  k_edge_acc<4, 2><<<g(EA), blk, 0, stream>>>(ei, E, N, xw2, as_, ad_, m_, dn_, h2);
  k_norm<4, 2><<<g((long)N * 8), blk, 0, stream>>>(h2, dn_, b2, N);

  // ---------------- pool + head ----------------
  k_pool<<<g(N), blk, 0, stream>>>(h2, pool, N);
  k_final<<<dim3(1), dim3(32), 0, stream>>>(pool, Wout, bout, out);
}